// backbone_pointnet2_90529320665195
// MI455X (gfx1250) — compile-verified
//
#include <hip/hip_runtime.h>

#define kB 8
#define kN 8192
#define NT 4   // 16-wide N tiles per wave (wave computes 16 x 64 output strip)

typedef __attribute__((ext_vector_type(2))) float v2f;
typedef __attribute__((ext_vector_type(8))) float v8f;

__device__ __forceinline__ int imin(int a, int b) { return a < b ? a : b; }

// ---------------------------------------------------------------------------
// Generic WMMA GEMM:  Y[b] = act( W (Cout x Cin)  x  X[b] (Cin x M) )
// X,Y are (B, C, M) contiguous.  act: 0=none, 1=relu(g*y+b), 2=leaky0.2(g*y+b)
// One wave computes a 16 x (16*NT) output strip with V_WMMA_F32_16X16X4_F32.
// Row overreads are clamped (not branched) and masked at the store.  The
// K-loop is branch-free with pointer-bumped coalesced loads; all NT B
// fragments are loaded into distinct registers BEFORE the WMMA burst so the
// compiler can clause the loads and stagger s_wait_loadcnt across the WMMAs.
// Requires M % (16*NT) == 0 (true for all call sites).
// ---------------------------------------------------------------------------
__global__ __launch_bounds__(32) void wmma_gemm_f32(
    const float* __restrict__ W, const float* __restrict__ X, float* __restrict__ Y,
    const float* __restrict__ gain, const float* __restrict__ bias,
    int Cin, int Cout, int M, int act)
{
  const int lane = threadIdx.x & 31;
  const int lr = lane & 15;          // N (for B/C) or M-row (for A)
  const int hi = lane >> 4;          // K-half selector
  const int n0  = blockIdx.x * (16 * NT) + lr;
  const int row = blockIdx.y * 16 + lr;
  const int rowc = imin(row, Cout - 1);          // clamp: masked at store
  const long b = blockIdx.z;
  const float* __restrict__ Xb = X + b * (long)Cin * M;
  float* __restrict__ Yb = Y + b * (long)Cout * M;
  const float* __restrict__ Wr = W + (long)rowc * Cin;

  v8f acc[NT];
  #pragma unroll
  for (int t = 0; t < NT; ++t) acc[t] = (v8f){0.f,0.f,0.f,0.f,0.f,0.f,0.f,0.f};

  const int Cin4 = Cin & ~3;
  const long step = 4L * M;                       // K-step advance in elements
  const float* wp  = Wr + 2 * hi;                 // A: K = k + 2*hi, k+2*hi+1
  const float* xp0 = Xb + (long)(2 * hi) * M + n0;
  const float* xp1 = xp0 + M;
  int k = 0;
  #pragma unroll 2
  for (; k < Cin4; k += 4) {
    v2f a;
    a.x = wp[0];
    a.y = wp[1];
    __builtin_prefetch(xp0 + 2 * step, 0, 1);     // speculative: safe if OOB
    v2f bbuf[NT];                                  // distinct regs -> load clause
    #pragma unroll
    for (int t = 0; t < NT; ++t) {
      bbuf[t].x = xp0[16 * t];
      bbuf[t].y = xp1[16 * t];
    }
    #pragma unroll
    for (int t = 0; t < NT; ++t) {
      acc[t] = __builtin_amdgcn_wmma_f32_16x16x4_f32(
          false, a, false, bbuf[t], (short)0, acc[t], false, false);
    }
    wp += 4; xp0 += step; xp1 += step;
  }
  if (k < Cin) {                      // K tail (Cin % 4 in {1,2,3}): zero A only
    const int ka = k + 2 * hi;
    const int ka0 = imin(ka, Cin - 1), ka1 = imin(ka + 1, Cin - 1);
    v2f a;
    float wx = Wr[ka0], wy = Wr[ka1];
    a.x = (ka     < Cin) ? wx : 0.0f;
    a.y = (ka + 1 < Cin) ? wy : 0.0f;
    const float* tp0 = Xb + (long)ka0 * M + n0;
    const float* tp1 = Xb + (long)ka1 * M + n0;
    v2f bbuf[NT];
    #pragma unroll
    for (int t = 0; t < NT; ++t) {
      bbuf[t].x = tp0[16 * t];         // clamped row; A==0 nullifies product
      bbuf[t].y = tp1[16 * t];
    }
    #pragma unroll
    for (int t = 0; t < NT; ++t) {
      acc[t] = __builtin_amdgcn_wmma_f32_16x16x4_f32(
          false, a, false, bbuf[t], (short)0, acc[t], false, false);
    }
  }

  #pragma unroll
  for (int r = 0; r < 8; ++r) {
    const int co = blockIdx.y * 16 + r + 8 * hi;  // C/D: VGPR r -> M=r | r+8
    if (co < Cout) {
      float gc = 1.0f, bc = 0.0f;
      if (act) { gc = gain[co]; bc = bias[co]; }
      float* yp = Yb + (long)co * M + n0;
      #pragma unroll
      for (int t = 0; t < NT; ++t) {
        float y = acc[t][r];
        if (act == 1)      { y = y * gc + bc; y = fmaxf(y, 0.0f); }
        else if (act == 2) { y = y * gc + bc; y = (y >= 0.0f) ? y : 0.2f * y; }
        yp[16 * t] = y;
      }
    }
  }
}

// ---------------------------------------------------------------------------
// Farthest point sampling: one block per batch; iterative argmax w/ LDS tree.
// ---------------------------------------------------------------------------
__global__ __launch_bounds__(256) void fps_kernel(
    const float* __restrict__ xyz, int n, int npoint, int* __restrict__ out_idx,
    float* __restrict__ dists)
{
  const int b = blockIdx.x;
  const float* p = xyz + (long)b * n * 3;
  float* dist = dists + (long)b * n;
  int* idx = out_idx + (long)b * npoint;
  __shared__ float s_val[256];
  __shared__ int   s_idx[256];
  __shared__ float s_last[3];
  const int t = threadIdx.x;
  for (int i = t; i < n; i += 256) dist[i] = 1e10f;
  int last = 0;
  if (t == 0) idx[0] = 0;
  __syncthreads();
  for (int it = 1; it < npoint; ++it) {
    if (t == 0) { s_last[0] = p[last*3]; s_last[1] = p[last*3+1]; s_last[2] = p[last*3+2]; }
    __syncthreads();
    const float lx = s_last[0], ly = s_last[1], lz = s_last[2];
    float best = -1.0f; int bidx = 0;
    for (int i = t; i < n; i += 256) {
      float dx = p[i*3] - lx, dy = p[i*3+1] - ly, dz = p[i*3+2] - lz;
      float d = dx*dx + dy*dy + dz*dz;
      float dm = fminf(dist[i], d);
      dist[i] = dm;
      if (dm > best) { best = dm; bidx = i; }
    }
    s_val[t] = best; s_idx[t] = bidx;
    __syncthreads();
    for (int s = 128; s > 0; s >>= 1) {
      if (t < s) {
        if (s_val[t+s] > s_val[t] || (s_val[t+s] == s_val[t] && s_idx[t+s] < s_idx[t])) {
          s_val[t] = s_val[t+s]; s_idx[t] = s_idx[t+s];
        }
      }
      __syncthreads();
    }
    last = s_idx[0];
    if (t == 0) idx[it] = last;
    __syncthreads();
  }
}

__global__ void gather_xyz_kernel(const float* __restrict__ xyz, const int* __restrict__ fidx,
                                  int n, int S, float* __restrict__ out)
{
  int g = blockIdx.x * blockDim.x + threadIdx.x;
  if (g >= kB * S) return;
  int b = g / S;
  int id = fidx[g];
  const float* src = xyz + ((long)b * n + id) * 3;
  out[g*3+0] = src[0]; out[g*3+1] = src[1]; out[g*3+2] = src[2];
}

// First-K-in-ball (index order), pad with first hit (or 0) — matches reference argsort trick.
__global__ void ball_query_kernel(const float* __restrict__ xyz, const float* __restrict__ nxyz,
                                  int n, int S, int K, float r2, int* __restrict__ out)
{
  int g = blockIdx.x * blockDim.x + threadIdx.x;
  if (g >= kB * S) return;
  int b = g / S;
  const float* p = xyz + (long)b * n * 3;
  float qx = nxyz[g*3], qy = nxyz[g*3+1], qz = nxyz[g*3+2];
  int* o = out + (long)g * K;
  int cnt = 0, first = 0;
  for (int i = 0; i < n && cnt < K; ++i) {
    float dx = p[i*3]-qx, dy = p[i*3+1]-qy, dz = p[i*3+2]-qz;
    if (dx*dx + dy*dy + dz*dz <= r2) { if (cnt == 0) first = i; o[cnt++] = i; }
  }
  for (int j = cnt; j < K; ++j) o[j] = first;
}

// grouped (B, 3+Cf, S, K): relative xyz + gathered feats (B,Cf,n)
__global__ void group_kernel(const float* __restrict__ xyz, const float* __restrict__ nxyz,
                             const float* __restrict__ feats, const int* __restrict__ idx,
                             int n, int S, int K, int Cf, float* __restrict__ out)
{
  long g = (long)blockIdx.x * blockDim.x + threadIdx.x;
  if (g >= (long)kB * S * K) return;
  int k = (int)(g % K); long t = g / K; int s = (int)(t % S); int b = (int)(t / S);
  int id = idx[g];
  long cs = (long)S * K;
  long base = (long)b * (3 + Cf) * cs + (long)s * K + k;
  const float* pp = xyz + ((long)b * n + id) * 3;
  const float* q  = nxyz + ((long)b * S + s) * 3;
  out[base + 0*cs] = pp[0] - q[0];
  out[base + 1*cs] = pp[1] - q[1];
  out[base + 2*cs] = pp[2] - q[2];
  const float* f = feats + (long)b * Cf * n + id;
  for (int c = 0; c < Cf; ++c) out[base + (long)(3 + c) * cs] = f[(long)c * n];
}

__global__ void maxk_kernel(const float* __restrict__ in, float* __restrict__ out,
                            int C, int S, int K)
{
  long g = (long)blockIdx.x * blockDim.x + threadIdx.x;
  if (g >= (long)kB * C * S) return;
  const float* p = in + g * K;
  float m = p[0];
  for (int k = 1; k < K; ++k) m = fmaxf(m, p[k]);
  out[g] = m;
}

// dst is (B, Cdst, M); copy src (B, C, M) into channel offset coff
__global__ void copy_ch_kernel(const float* __restrict__ src, float* __restrict__ dst,
                               int C, int M, int Cdst, int coff)
{
  long g = (long)blockIdx.x * blockDim.x + threadIdx.x;
  if (g >= (long)kB * C * M) return;
  int m = (int)(g % M); long t = g / M; int c = (int)(t % C); int b = (int)(t / C);
  dst[((long)b * Cdst + coff + c) * M + m] = src[g];
}

// write xyz (B, M, 3) transposed into channels [coff, coff+3)
__global__ void xyzT_ch_kernel(const float* __restrict__ xyz, float* __restrict__ dst,
                               int M, int Cdst, int coff)
{
  long g = (long)blockIdx.x * blockDim.x + threadIdx.x;
  if (g >= (long)kB * 3 * M) return;
  int m = (int)(g % M); long t = g / M; int c = (int)(t % 3); int b = (int)(t / 3);
  dst[((long)b * Cdst + coff + c) * M + m] = xyz[((long)b * M + m) * 3 + c];
}

// broadcast src (B, C) to channels [coff, coff+C) across M columns
__global__ void bcast_ch_kernel(const float* __restrict__ src, float* __restrict__ dst,
                                int C, int M, int Cdst, int coff)
{
  long g = (long)blockIdx.x * blockDim.x + threadIdx.x;
  if (g >= (long)kB * C * M) return;
  int m = (int)(g % M); long t = g / M; int c = (int)(t % C); int b = (int)(t / C);
  dst[((long)b * Cdst + coff + c) * M + m] = src[(long)b * C + c];
}

// 3-NN search + inverse-distance weights
__global__ void knn3_kernel(const float* __restrict__ unk, const float* __restrict__ kn,
                            int nU, int nK, int* __restrict__ oidx, float* __restrict__ ow)
{
  int g = blockIdx.x * blockDim.x + threadIdx.x;
  if (g >= kB * nU) return;
  int b = g / nU;
  float ux = unk[g*3], uy = unk[g*3+1], uz = unk[g*3+2];
  const float* p = kn + (long)b * nK * 3;
  float d0 = 1e30f, d1 = 1e30f, d2 = 1e30f;
  int i0 = 0, i1 = 0, i2 = 0;
  for (int i = 0; i < nK; ++i) {
    float dx = p[i*3]-ux, dy = p[i*3+1]-uy, dz = p[i*3+2]-uz;
    float d = dx*dx + dy*dy + dz*dz;
    if (d < d0)      { d2=d1;i2=i1; d1=d0;i1=i0; d0=d;i0=i; }
    else if (d < d1) { d2=d1;i2=i1; d1=d;i1=i; }
    else if (d < d2) { d2=d;i2=i; }
  }
  float w0 = 1.0f/(d0+1e-8f), w1 = 1.0f/(d1+1e-8f), w2 = 1.0f/(d2+1e-8f);
  float s = w0 + w1 + w2;
  oidx[g*3+0]=i0; oidx[g*3+1]=i1; oidx[g*3+2]=i2;
  ow[g*3+0]=w0/s; ow[g*3+1]=w1/s; ow[g*3+2]=w2/s;
}

// interp: dst[b, coff+c, m] = sum_j w[j] * kf[b, c, idx[j]]
__global__ void interp_ch_kernel(const float* __restrict__ kf, const int* __restrict__ idx3,
                                 const float* __restrict__ w3, int C, int nU, int nK,
                                 float* __restrict__ dst, int Cdst, int coff)
{
  long g = (long)blockIdx.x * blockDim.x + threadIdx.x;
  if (g >= (long)kB * C * nU) return;
  int m = (int)(g % nU); long t = g / nU; int c = (int)(t % C); int b = (int)(t / C);
  const int*   id = idx3 + ((long)b * nU + m) * 3;
  const float* w  = w3   + ((long)b * nU + m) * 3;
  const float* f  = kf + ((long)b * C + c) * nK;
  dst[((long)b * Cdst + coff + c) * nU + m] = w[0]*f[id[0]] + w[1]*f[id[1]] + w[2]*f[id[2]];
}

// final: rearrange afford (B,2,N) and grasp (B,135,N) into output tuple layout
__global__ void writeout_kernel(const float* __restrict__ afford, const float* __restrict__ grasp,
                                float* __restrict__ out)
{
  int g = blockIdx.x * blockDim.x + threadIdx.x;
  if (g >= kB * kN) return;
  int b = g / kN, n = g % kN;
  const long MN = kN;
  float* oA = out + (long)g * 2;
  oA[0] = afford[((long)b*2 + 0)*MN + n];
  oA[1] = afford[((long)b*2 + 1)*MN + n];
  const float* gb = grasp + (long)b * 135 * MN + n;
  float* oG = out + 131072L + (long)g * 10;
  for (int gc = 0; gc < 2; ++gc)
    for (int t = 0; t < 5; ++t) oG[gc*5+t] = gb[(long)(gc*5+t)*MN];
  float* oP = out + 786432L + (long)g * 25;
  for (int gc = 0; gc < 5; ++gc)
    for (int t = 0; t < 5; ++t) oP[gc*5+t] = gb[(long)((gc+2)*5+t)*MN];
  float* oJ = out + 2424832L + (long)g * 100;
  for (int gc = 0; gc < 20; ++gc)
    for (int t = 0; t < 5; ++t) {
      float v = gb[(long)((gc+7)*5+t)*MN];
      oJ[gc*5+t] = fminf(fmaxf(v, -0.0001f), 1.0001f);
    }
}

// ---------------------------------------------------------------------------
static inline unsigned cdivu(long a, long b) { return (unsigned)((a + b - 1) / b); }

extern "C" void kernel_launch(void* const* d_in, const int* in_sizes, int n_in,
                              void* d_out, int out_size, void* d_ws, size_t ws_size,
                              hipStream_t stream)
{
  (void)in_sizes; (void)n_in; (void)out_size; (void)ws_size;
  const float* xyz    = (const float*)d_in[0];
  const float* points = (const float*)d_in[1];
  const float* P[70];
  for (int i = 0; i < 70; ++i) P[i] = (const float*)d_in[i];
  float* out = (float*)d_out;

  // --- workspace arena (floats / ints, 4B aligned) ---
  size_t off = 0;
  char* base = (char*)d_ws;
  auto af = [&](size_t nf) { float* p = (float*)(base + off); off += nf * 4; return p; };
  auto ai = [&](size_t ni) { int*   p = (int*)  (base + off); off += ni * 4; return p; };
  float* dists  = af((size_t)kB * kN);
  int*   fidx   = ai((size_t)kB * 1024);
  int*   idxb   = ai((size_t)kB * 1024 * 32);
  int*   knn_i  = ai((size_t)kB * kN * 3);
  float* knn_w  = af((size_t)kB * kN * 3);
  float* l1_xyz = af((size_t)kB * 1024 * 3);
  float* l2_xyz = af((size_t)kB * 256 * 3);
  float* l3_xyz = af((size_t)kB * 64 * 3);
  float* l1_f   = af((size_t)kB * 64 * 1024);
  float* l2_f   = af((size_t)kB * 128 * 256);
  float* l3_f   = af((size_t)kB * 256 * 64);
  float* l4_f   = af((size_t)kB * 512);
  float* l3_f2  = af((size_t)kB * 256 * 64);
  float* l2_f2  = af((size_t)kB * 256 * 256);
  float* l1_f2  = af((size_t)kB * 128 * 1024);
  float* SCR0   = af(8912896);   // max: (B,136,8192)
  float* SCR1   = af(8388608);   // max: (B,128,8192)
  float* SCR2   = af(16777216);  // max: (B,128,16384)

  auto gemm = [&](const float* W, const float* X, float* Y, const float* g, const float* b,
                  int Cin, int Cout, int M, int act) {
    dim3 grid((unsigned)(M / (16 * NT)), cdivu(Cout, 16), kB);
    hipLaunchKernelGGL(wmma_gemm_f32, grid, dim3(32), 0, stream, W, X, Y, g, b, Cin, Cout, M, act);
  };
  const unsigned T = 256;

  // ===== SA1: N=8192 -> S=1024, K=32, mlp 6->32->32->64 =====
  hipLaunchKernelGGL(fps_kernel, dim3(kB), dim3(T), 0, stream, xyz, kN, 1024, fidx, dists);
  hipLaunchKernelGGL(gather_xyz_kernel, dim3(cdivu((long)kB*1024, T)), dim3(T), 0, stream, xyz, fidx, kN, 1024, l1_xyz);
  hipLaunchKernelGGL(ball_query_kernel, dim3(cdivu((long)kB*1024, T)), dim3(T), 0, stream, xyz, l1_xyz, kN, 1024, 32, 0.01f, idxb);
  hipLaunchKernelGGL(group_kernel, dim3(cdivu((long)kB*1024*32, T)), dim3(T), 0, stream, xyz, l1_xyz, points, idxb, kN, 1024, 32, 3, SCR0);
  gemm(P[2], SCR0, SCR1, P[3], P[4], 6, 32, 32768, 1);
  gemm(P[5], SCR1, SCR0, P[6], P[7], 32, 32, 32768, 1);
  gemm(P[8], SCR0, SCR2, P[9], P[10], 32, 64, 32768, 1);
  hipLaunchKernelGGL(maxk_kernel, dim3(cdivu((long)kB*64*1024, T)), dim3(T), 0, stream, SCR2, l1_f, 64, 1024, 32);

  // ===== SA2: 1024 -> S=256, K=64, mlp 67->64->64->128 =====
  hipLaunchKernelGGL(fps_kernel, dim3(kB), dim3(T), 0, stream, l1_xyz, 1024, 256, fidx, dists);
  hipLaunchKernelGGL(gather_xyz_kernel, dim3(cdivu((long)kB*256, T)), dim3(T), 0, stream, l1_xyz, fidx, 1024, 256, l2_xyz);
  hipLaunchKernelGGL(ball_query_kernel, dim3(cdivu((long)kB*256, T)), dim3(T), 0, stream, l1_xyz, l2_xyz, 1024, 256, 64, 0.04f, idxb);
  hipLaunchKernelGGL(group_kernel, dim3(cdivu((long)kB*256*64, T)), dim3(T), 0, stream, l1_xyz, l2_xyz, l1_f, idxb, 1024, 256, 64, 64, SCR0);
  gemm(P[11], SCR0, SCR1, P[12], P[13], 67, 64, 16384, 1);
  gemm(P[14], SCR1, SCR0, P[15], P[16], 64, 64, 16384, 1);
  gemm(P[17], SCR0, SCR2, P[18], P[19], 64, 128, 16384, 1);
  hipLaunchKernelGGL(maxk_kernel, dim3(cdivu((long)kB*128*256, T)), dim3(T), 0, stream, SCR2, l2_f, 128, 256, 64);

  // ===== SA3: 256 -> S=64, K=128, mlp 131->128->128->256 =====
  hipLaunchKernelGGL(fps_kernel, dim3(kB), dim3(T), 0, stream, l2_xyz, 256, 64, fidx, dists);
  hipLaunchKernelGGL(gather_xyz_kernel, dim3(cdivu((long)kB*64, T)), dim3(T), 0, stream, l2_xyz, fidx, 256, 64, l3_xyz);
  hipLaunchKernelGGL(ball_query_kernel, dim3(cdivu((long)kB*64, T)), dim3(T), 0, stream, l2_xyz, l3_xyz, 256, 64, 128, 0.16f, idxb);
  hipLaunchKernelGGL(group_kernel, dim3(cdivu((long)kB*64*128, T)), dim3(T), 0, stream, l2_xyz, l3_xyz, l2_f, idxb, 256, 64, 128, 128, SCR0);
  gemm(P[20], SCR0, SCR1, P[21], P[22], 131, 128, 8192, 1);
  gemm(P[23], SCR1, SCR0, P[24], P[25], 128, 128, 8192, 1);
  gemm(P[26], SCR0, SCR2, P[27], P[28], 128, 256, 8192, 1);
  hipLaunchKernelGGL(maxk_kernel, dim3(cdivu((long)kB*256*64, T)), dim3(T), 0, stream, SCR2, l3_f, 256, 64, 128);

  // ===== SA4 (group-all): (B,259,64), mlp 259->256->256->512, max over 64 =====
  hipLaunchKernelGGL(xyzT_ch_kernel, dim3(cdivu((long)kB*3*64, T)), dim3(T), 0, stream, l3_xyz, SCR0, 64, 259, 0);
  hipLaunchKernelGGL(copy_ch_kernel, dim3(cdivu((long)kB*256*64, T)), dim3(T), 0, stream, l3_f, SCR0, 256, 64, 259, 3);
  gemm(P[29], SCR0, SCR1, P[30], P[31], 259, 256, 64, 1);
  gemm(P[32], SCR1, SCR0, P[33], P[34], 256, 256, 64, 1);
  gemm(P[35], SCR0, SCR2, P[36], P[37], 256, 512, 64, 1);
  hipLaunchKernelGGL(maxk_kernel, dim3(cdivu((long)kB*512, T)), dim3(T), 0, stream, SCR2, l4_f, 512, 1, 64);

  // ===== FP4: broadcast l4_f(512) + l3_f(256) -> mlp 768->256->256 =====
  hipLaunchKernelGGL(bcast_ch_kernel, dim3(cdivu((long)kB*512*64, T)), dim3(T), 0, stream, l4_f, SCR0, 512, 64, 768, 0);
  hipLaunchKernelGGL(copy_ch_kernel, dim3(cdivu((long)kB*256*64, T)), dim3(T), 0, stream, l3_f, SCR0, 256, 64, 768, 512);
  gemm(P[38], SCR0, SCR1, P[39], P[40], 768, 256, 64, 1);
  gemm(P[41], SCR1, l3_f2, P[42], P[43], 256, 256, 64, 1);

  // ===== FP3: interp l3_f2 (256ch) onto 256 pts + l2_f(128) -> mlp 384->256->256 =====
  hipLaunchKernelGGL(knn3_kernel, dim3(cdivu((long)kB*256, T)), dim3(T), 0, stream, l2_xyz, l3_xyz, 256, 64, knn_i, knn_w);
  hipLaunchKernelGGL(interp_ch_kernel, dim3(cdivu((long)kB*256*256, T)), dim3(T), 0, stream, l3_f2, knn_i, knn_w, 256, 256, 64, SCR0, 384, 0);
  hipLaunchKernelGGL(copy_ch_kernel, dim3(cdivu((long)kB*128*256, T)), dim3(T), 0, stream, l2_f, SCR0, 128, 256, 384, 256);
  gemm(P[44], SCR0, SCR1, P[45], P[46], 384, 256, 256, 1);
  gemm(P[47], SCR1, l2_f2, P[48], P[49], 256, 256, 256, 1);

  // ===== FP2: interp l2_f2 (256ch) onto 1024 pts + l1_f(64) -> mlp 320->256->128 =====
  hipLaunchKernelGGL(knn3_kernel, dim3(cdivu((long)kB*1024, T)), dim3(T), 0, stream, l1_xyz, l2_xyz, 1024, 256, knn_i, knn_w);
  hipLaunchKernelGGL(interp_ch_kernel, dim3(cdivu((long)kB*256*1024, T)), dim3(T), 0, stream, l2_f2, knn_i, knn_w, 256, 1024, 256, SCR0, 320, 0);
  hipLaunchKernelGGL(copy_ch_kernel, dim3(cdivu((long)kB*64*1024, T)), dim3(T), 0, stream, l1_f, SCR0, 64, 1024, 320, 256);
  gemm(P[50], SCR0, SCR1, P[51], P[52], 320, 256, 1024, 1);
  gemm(P[53], SCR1, l1_f2, P[54], P[55], 256, 128, 1024, 1);

  // ===== FP1: interp l1_f2 (128ch) onto 8192 pts + skip(xyz^T, points) -> mlp 134->128->128 =====
  hipLaunchKernelGGL(knn3_kernel, dim3(cdivu((long)kB*kN, T)), dim3(T), 0, stream, xyz, l1_xyz, kN, 1024, knn_i, knn_w);
  hipLaunchKernelGGL(interp_ch_kernel, dim3(cdivu((long)kB*128*kN, T)), dim3(T), 0, stream, l1_f2, knn_i, knn_w, 128, kN, 1024, SCR0, 134, 0);
  hipLaunchKernelGGL(xyzT_ch_kernel, dim3(cdivu((long)kB*3*kN, T)), dim3(T), 0, stream, xyz, SCR0, kN, 134, 128);
  hipLaunchKernelGGL(copy_ch_kernel, dim3(cdivu((long)kB*3*kN, T)), dim3(T), 0, stream, points, SCR0, 3, kN, 134, 131);
  gemm(P[56], SCR0, SCR1, P[57], P[58], 134, 128, kN, 1);
  gemm(P[59], SCR1, SCR0, P[60], P[61], 128, 128, kN, 1);   // feat -> SCR0

  // ===== Head: leaky(w1), leaky(w2), w3 (135ch), w4 (2ch), rearrange =====
  gemm(P[62], SCR0, SCR1, P[66], P[67], 128, 128, kN, 2);
  gemm(P[63], SCR1, SCR0, P[68], P[69], 128, 128, kN, 2);
  gemm(P[64], SCR0, SCR2, nullptr, nullptr, 128, 135, kN, 0);  // grasp
  gemm(P[65], SCR0, SCR1, nullptr, nullptr, 128, 2,   kN, 0);  // afford
  hipLaunchKernelGGL(writeout_kernel, dim3(cdivu((long)kB*kN, T)), dim3(T), 0, stream, SCR1, SCR2, out);
}